// Sparsemax_1271310320382
// MI455X (gfx1250) — compile-verified
//
#include <hip/hip_runtime.h>

#define D    4096
#define TPB  256
#define NCH  4          // 4 async chunks of 4 KB each; thread t owns 16B per chunk

typedef float v4f __attribute__((ext_vector_type(4)));

__global__ __launch_bounds__(TPB) void sparsemax_kernel(const float* __restrict__ x,
                                                        float* __restrict__ out) {
    __shared__ __align__(16) float s_x[D];      // staged row (async global -> LDS)
    __shared__ __align__(16) float s_cand[D];   // compacted candidates (y > -1)
    __shared__ float s_wmax[8];
    __shared__ int   s_wtot[8];
    __shared__ float s_tau;

    const int      row  = blockIdx.x;
    const int      t    = threadIdx.x;
    const unsigned lane = (unsigned)t & 31u;
    const unsigned wid  = (unsigned)t >> 5;

    const float* xrow = x + (size_t)row * D;

    // ---- Stage row into LDS with async loads, GVS mode (ASYNCcnt-tracked) ----
    // the imm offset is added to both the LDS address (VDST) and the global address (SADDR + VADDR)
    // so one 32-bit voffset covers both sides; chunk stride via offset: imm.
    // Streaming input, zero reuse (512 MB stream vs 192 MB L2) -> th:TH_LOAD_NT.
    const unsigned lds_base = (unsigned)(size_t)(void*)s_x;   // low 32 bits = LDS offset
    {
        const unsigned ldsaddr = lds_base + (unsigned)(t * 16);
        const unsigned voff    = (unsigned)(t * 16);
        const unsigned long long sbase = (unsigned long long)(const void*)xrow;
        asm volatile("global_load_async_to_lds_b128 %0, %1, %2 th:TH_LOAD_NT"
                     :: "v"(ldsaddr), "v"(voff), "s"(sbase) : "memory");
        asm volatile("global_load_async_to_lds_b128 %0, %1, %2 offset:4096 th:TH_LOAD_NT"
                     :: "v"(ldsaddr), "v"(voff), "s"(sbase) : "memory");
        asm volatile("global_load_async_to_lds_b128 %0, %1, %2 offset:8192 th:TH_LOAD_NT"
                     :: "v"(ldsaddr), "v"(voff), "s"(sbase) : "memory");
        asm volatile("global_load_async_to_lds_b128 %0, %1, %2 offset:12288 th:TH_LOAD_NT"
                     :: "v"(ldsaddr), "v"(voff), "s"(sbase) : "memory");
    }
    asm volatile("s_wait_asynccnt 0x0" ::: "memory");

    // Each thread reads back exactly the 16 floats its own wave staged.
    v4f v[NCH];
#pragma unroll
    for (int c = 0; c < NCH; ++c)
        v[c] = *reinterpret_cast<const v4f*>(&s_x[c * 1024 + t * 4]);

    // ---- Row max (wave32 butterfly + 8-wave LDS combine) ----
    float m = v[0][0];
#pragma unroll
    for (int c = 0; c < NCH; ++c)
        m = fmaxf(m, fmaxf(fmaxf(v[c][0], v[c][1]), fmaxf(v[c][2], v[c][3])));
#pragma unroll
    for (int off = 16; off >= 1; off >>= 1)
        m = fmaxf(m, __shfl_xor(m, off, 32));
    if (lane == 0) s_wmax[wid] = m;
    __syncthreads();
    float rowmax = s_wmax[0];
#pragma unroll
    for (int i = 1; i < 8; ++i) rowmax = fmaxf(rowmax, s_wmax[i]);

    // ---- Count candidates: only y = x - rowmax > -1 can be in the support ----
    float e[16];
#pragma unroll
    for (int c = 0; c < NCH; ++c) {
        e[4 * c + 0] = v[c][0]; e[4 * c + 1] = v[c][1];
        e[4 * c + 2] = v[c][2]; e[4 * c + 3] = v[c][3];
    }
    int nc = 0;
#pragma unroll
    for (int j = 0; j < 16; ++j)
        nc += (e[j] - rowmax > -1.0f) ? 1 : 0;

    // Deterministic compaction: wave inclusive scan (shfl_up) + per-wave bases.
    int scan = nc;
#pragma unroll
    for (int off = 1; off < 32; off <<= 1) {
        int n = __shfl_up(scan, off, 32);
        if (lane >= (unsigned)off) scan += n;
    }
    if (lane == 31) s_wtot[wid] = scan;       // wave total (inclusive @ lane 31)
    int laneExcl = scan - nc;
    __syncthreads();
    int base = 0, totalAll = 0;
#pragma unroll
    for (int i = 0; i < 8; ++i) {
        int w = s_wtot[i];
        if (i < (int)wid) base += w;
        totalAll += w;
    }
    int wpos = base + laneExcl;
#pragma unroll
    for (int j = 0; j < 16; ++j) {
        float y = e[j] - rowmax;
        if (y > -1.0f) s_cand[wpos++] = y;
    }
    __syncthreads();

    // ---- Newton on the candidate set (wave 0 only; wave-level reductions) ----
    // f(tau) = sum(relu(y - tau)) - 1 is piecewise linear, convex, decreasing;
    // Newton from tau = -1 converges monotonically to the exact sparsemax tau.
    if (wid == 0) {
        float tau = -1.0f;
        for (int it = 0; it < 48; ++it) {
            float S = 0.0f;
            int   k = 0;
            for (int j = (int)lane; j < totalAll; j += 32) {
                float d = s_cand[j] - tau;
                if (d > 0.0f) { S += d; k += 1; }
            }
#pragma unroll
            for (int off = 16; off >= 1; off >>= 1) {
                S += __shfl_xor(S, off, 32);
                k += __shfl_xor(k, off, 32);
            }
            if (k == 0) break;
            float tn = tau + (S - 1.0f) / (float)k;
            if (!(tn > tau)) break;   // converged (monotone ascent)
            tau = tn;
        }
        if (lane == 0) s_tau = tau;
    }
    __syncthreads();

    // ---- Output: out = max(x - (rowmax + tau), 0); streaming NT b128 stores ----
    const float thr = rowmax + s_tau;
    float* orow = out + (size_t)row * D;
#pragma unroll
    for (int c = 0; c < NCH; ++c) {
        v4f o;
        o[0] = fmaxf(e[4 * c + 0] - thr, 0.0f);
        o[1] = fmaxf(e[4 * c + 1] - thr, 0.0f);
        o[2] = fmaxf(e[4 * c + 2] - thr, 0.0f);
        o[3] = fmaxf(e[4 * c + 3] - thr, 0.0f);
        __builtin_nontemporal_store(o, reinterpret_cast<v4f*>(&orow[c * 1024 + t * 4]));
    }
}

extern "C" void kernel_launch(void* const* d_in, const int* in_sizes, int n_in,
                              void* d_out, int out_size, void* d_ws, size_t ws_size,
                              hipStream_t stream) {
    const float* x   = (const float*)d_in[0];
    float*       out = (float*)d_out;
    const int rows = in_sizes[0] / D;   // 8 * 2048 = 16384
    hipLaunchKernelGGL(sparsemax_kernel, dim3(rows), dim3(TPB), 0, stream, x, out);
}